// ROMModel_3118146257101
// MI455X (gfx1250) — compile-verified
//
#include <hip/hip_runtime.h>
#include <hip/hip_bf16.h>

#define T_DIM   8192
#define N_DIM   4096
#define L       64
#define P       32
#define INDIM   192
#define CHUNK   64
#define NCHUNK  (T_DIM / CHUNK)   // 128

// Workspace layout (float offsets). Total ~540,800 floats (~2.2 MB).
#define WS_SUMD   0        // (Y - Xrec)^2 accumulator
#define WS_SUMY   1        // Y^2 accumulator
#define WS_SEL    16       // 32 ints (clipped selected idx)
#define WS_M      64       // M = A + phi_bar*S   (64x64)
#define WS_M64    4160     // M^64                (64x64)
#define WS_XH0    8256     // xh0                 (64)
#define WS_B      8320     // chunk boundaries    (128x64)
#define WS_XHSEQ  16512    // xh_seq              (8192x64)

typedef float v2f __attribute__((ext_vector_type(2)));
typedef float v8f __attribute__((ext_vector_type(8)));

// ---------------- Stage 0: temp, argmax selection ----------------
__global__ void k_setup(const float* __restrict__ temp,
                        const float* __restrict__ logits,
                        float* __restrict__ ws,
                        float* __restrict__ out_tail) {
    int t = threadIdx.x;
    if (t == 0) {
        ws[WS_SUMD] = 0.f;
        ws[WS_SUMY] = 0.f;
        float nt = fmaxf(0.01f, temp[0] * 0.999f);
        out_tail[1] = nt;                       // new_temp
    }
    if (t < P) {
        const float* row = logits + t * INDIM;
        int   best = 0;
        float bv   = row[0];
        for (int j = 1; j < INDIM; ++j) {       // first-max tiebreak == jnp.argmax
            float v = row[j];
            if (v > bv) { bv = v; best = j; }
        }
        out_tail[2 + t] = (float)best;          // raw selected_idx (output)
        ((int*)ws)[WS_SEL + t] = best > (L - 1) ? (L - 1) : best; // clipped (gather)
    }
}

// ---------------- Stage 1: M = A + phi_bar @ S ----------------
__global__ void k_build_M(const float* __restrict__ A,
                          const float* __restrict__ phi_bar_t,   // (P, L)
                          float* __restrict__ ws) {
    __shared__ float rows[L][L];
    int i = threadIdx.x;                        // row 0..63
    const int* sel = (const int*)ws + WS_SEL;
    for (int l = 0; l < L; ++l) rows[i][l] = A[i * L + l];
    for (int p = 0; p < P; ++p) rows[i][sel[p]] += phi_bar_t[p * L + i];
    float* M = ws + WS_M;
    for (int l = 0; l < L; ++l) M[i * L + l] = rows[i][l];
}

// ---------------- Stage 2: xh0 = U_l^T @ X_train[0] ----------------
__global__ void k_xh0(const float* __restrict__ U_l,     // (N, L)
                      const float* __restrict__ X_train, // row 0 used
                      float* __restrict__ ws) {
    __shared__ float red[256];
    int l = blockIdx.x, t = threadIdx.x;
    float s = 0.f;
    for (int n = t; n < N_DIM; n += 256) s += U_l[n * L + l] * X_train[n];
    red[t] = s;
    __syncthreads();
    for (int k = 128; k > 0; k >>= 1) {
        if (t < k) red[t] += red[t + k];
        __syncthreads();
    }
    if (t == 0) ws[WS_XH0 + l] = red[0];
}

// ---------------- Stage 3: M^64 by squaring + 128 boundary states ----------------
__global__ void k_bounds(float* __restrict__ ws) {
    __shared__ float bufA[L * L], bufB[L * L];
    __shared__ float xv[L], xn[L];
    int t = threadIdx.x;                        // 256 threads
    float* M = ws + WS_M;
    for (int i = t; i < L * L; i += 256) bufA[i] = M[i];
    __syncthreads();
    float* cur = bufA;
    float* nxt = bufB;
    for (int it = 0; it < 6; ++it) {            // M^2, M^4, ..., M^64
        for (int idx = t; idx < L * L; idx += 256) {
            int r = idx >> 6, c = idx & 63;
            float s = 0.f;
            for (int k = 0; k < L; ++k) s += cur[r * L + k] * cur[k * L + c];
            nxt[idx] = s;
        }
        __syncthreads();
        float* tmp = cur; cur = nxt; nxt = tmp;
    }
    float* M64 = ws + WS_M64;
    for (int i = t; i < L * L; i += 256) M64[i] = cur[i];
    // serial boundary chain: b_c = (M^64)^c @ xh0
    if (t < L) xv[t] = ws[WS_XH0 + t];
    __syncthreads();
    float* B = ws + WS_B;
    if (t < L) B[t] = xv[t];                    // b_0 = xh0
    for (int c = 1; c < NCHUNK; ++c) {
        if (t < L) {
            float s = 0.f;
            for (int k = 0; k < L; ++k) s += cur[t * L + k] * xv[k];
            xn[t] = s;
        }
        __syncthreads();
        if (t < L) { xv[t] = xn[t]; B[c * L + t] = xn[t]; }
        __syncthreads();
    }
}

// ---------------- Stage 4: parallel chunk expansion (128 independent scans) ----------------
__global__ void __launch_bounds__(64) k_chunks(float* __restrict__ ws) {
    __shared__ float x[L];
    int c = blockIdx.x, i = threadIdx.x;        // 64 threads, thread i owns row i of M
    const float* M = ws + WS_M;
    float m[L];
#pragma unroll
    for (int l = 0; l < L; ++l) m[l] = M[i * L + l];
    x[i] = ws[WS_B + c * L + i];
    __syncthreads();
    float* xh = ws + WS_XHSEQ + (size_t)c * CHUNK * L;
    for (int j = 0; j < CHUNK; ++j) {
        float s = 0.f;
#pragma unroll
        for (int l = 0; l < L; ++l) s += m[l] * x[l];
        __syncthreads();
        x[i] = s;
        xh[j * L + i] = s;                      // xh_seq[c*64 + j][i]
        __syncthreads();
    }
}

// ---------------- Stage 5: X_rec = xh_seq @ U_l^T via V_WMMA_F32_16X16X4_F32,
//                  fused ||Y - Xrec||^2 and ||Y||^2 partial sums ----------------
__global__ void __launch_bounds__(256) k_gemm(const float* __restrict__ xh_seq, // (T, 64)
                                              const float* __restrict__ U_l,    // (N, 64)
                                              const float* __restrict__ Y,      // (T, N)
                                              float* __restrict__ Xrec,
                                              float* __restrict__ ws_acc) {
    const int lane = threadIdx.x & 31;
    const int wave = threadIdx.x >> 5;          // 8 waves
    const int m0   = blockIdx.x * 16;
    const int n0   = blockIdx.y * 128 + wave * 16;
    const int half = lane >> 4;                 // K sub-pair select (lanes 16-31 -> K+2)
    const int l15  = lane & 15;

    const float* Arow = xh_seq + (size_t)(m0 + l15) * L;  // A fragment: row m0+l15
    const float* Brow = U_l    + (size_t)(n0 + l15) * L;  // B[k][n] = U_l[n*64+k]

    v8f c = {0.f, 0.f, 0.f, 0.f, 0.f, 0.f, 0.f, 0.f};
#pragma unroll
    for (int kk = 0; kk < 16; ++kk) {
        const int kb = kk * 4 + half * 2;       // 8B-aligned
        v2f a = *(const v2f*)(Arow + kb);
        v2f b = *(const v2f*)(Brow + kb);
        c = __builtin_amdgcn_wmma_f32_16x16x4_f32(false, a, false, b,
                                                  (short)0, c, false, false);
    }

    // C/D layout: VGPR r -> M = m0 + r + 8*half, N = n0 + (lane&15)
    float sd = 0.f, sy = 0.f;
    const int col = n0 + l15;
#pragma unroll
    for (int r = 0; r < 8; ++r) {
        const size_t off = (size_t)(m0 + r + half * 8) * N_DIM + col;
        float y = Y[off];
        float v = c[r];
        Xrec[off] = v;
        float d = y - v;
        sd += d * d;
        sy += y * y;
    }

    __shared__ float rd[256], ry[256];
    rd[threadIdx.x] = sd;
    ry[threadIdx.x] = sy;
    __syncthreads();
    for (int k = 128; k > 0; k >>= 1) {
        if (threadIdx.x < k) {
            rd[threadIdx.x] += rd[threadIdx.x + k];
            ry[threadIdx.x] += ry[threadIdx.x + k];
        }
        __syncthreads();
    }
    if (threadIdx.x == 0) {
        atomicAdd(&ws_acc[WS_SUMD], rd[0]);
        atomicAdd(&ws_acc[WS_SUMY], ry[0]);
    }
}

// ---------------- Stage 6: err = sqrt(sum_d2 / sum_y2) ----------------
__global__ void k_final(const float* __restrict__ ws, float* __restrict__ out_tail) {
    if (threadIdx.x == 0) out_tail[0] = sqrtf(ws[WS_SUMD] / ws[WS_SUMY]);
}

extern "C" void kernel_launch(void* const* d_in, const int* in_sizes, int n_in,
                              void* d_out, int out_size, void* d_ws, size_t ws_size,
                              hipStream_t stream) {
    // setup_inputs() order:
    // 0: X_tilde (unused)  1: X_train  2: Y_train  3: temp  4: phi_bar_t
    // 5: A_tilde  6: U_l  7: logits  8: gumbel_g (unused)
    const float* X_train  = (const float*)d_in[1];
    const float* Y_train  = (const float*)d_in[2];
    const float* temp     = (const float*)d_in[3];
    const float* phi_bar  = (const float*)d_in[4];
    const float* A_tilde  = (const float*)d_in[5];
    const float* U_l      = (const float*)d_in[6];
    const float* logits   = (const float*)d_in[7];

    float* ws  = (float*)d_ws;
    float* out = (float*)d_out;
    float* out_tail = out + (size_t)T_DIM * N_DIM;   // [err, new_temp, sel[32]]

    k_setup  <<<1, 64, 0, stream>>>(temp, logits, ws, out_tail);
    k_build_M<<<1, L, 0, stream>>>(A_tilde, phi_bar, ws);
    k_xh0    <<<L, 256, 0, stream>>>(U_l, X_train, ws);
    k_bounds <<<1, 256, 0, stream>>>(ws);
    k_chunks <<<NCHUNK, 64, 0, stream>>>(ws);
    dim3 grid(T_DIM / 16, N_DIM / 128);              // 512 x 32 blocks, 8 waves each
    k_gemm   <<<grid, 256, 0, stream>>>(ws + WS_XHSEQ, U_l, Y_train, out, ws);
    k_final  <<<1, 32, 0, stream>>>(ws, out_tail);
}